// userShoppingGCNModel_25168508355091
// MI455X (gfx1250) — compile-verified
//
#include <hip/hip_runtime.h>
#include <stdint.h>

#define N_USERS 50000
#define N_PRODS 50000
#define N_NODES 100000
#define N_EDGES 600000
#define UD 64
#define PD 100
#define HD 128
#define OD 16

typedef float v2f __attribute__((ext_vector_type(2)));
typedef float v8f __attribute__((ext_vector_type(8)));

// ---------------------------------------------------------------------------
// Degree / normalization kernels
// ---------------------------------------------------------------------------
__global__ void k_init_deg(float* deg) {
    int i = blockIdx.x * 256 + threadIdx.x;
    if (i < N_NODES) deg[i] = 1.0f;   // self-loop contributes 1
}

__global__ void k_deg_edges(const long long* __restrict__ dst, float* deg) {
    int e = blockIdx.x * 256 + threadIdx.x;
    if (e < N_EDGES) unsafeAtomicAdd(&deg[(int)dst[e]], 1.0f);
}

__global__ void k_dinv(float* __restrict__ dinv, const float* __restrict__ deg) {
    int i = blockIdx.x * 256 + threadIdx.x;
    if (i < N_NODES) dinv[i] = rsqrtf(deg[i]);  // deg >= 1 always
}

// ---------------------------------------------------------------------------
// WMMA f32 GEMM:  C[M, NTILES*16] = (relu?)A[M,K] @ W[K, NTILES*16] (+ bias)
// One wave computes a 16-row x (NTILES*16)-col stripe; A read once per GEMM.
// ---------------------------------------------------------------------------
template <int NTILES, bool RELU, bool BIAS>
__global__ __launch_bounds__(128) void gemm_wmma(
    const float* __restrict__ A, const float* __restrict__ W,
    const float* __restrict__ bias, float* __restrict__ C,
    int K, int mtiles)
{
    constexpr int N = NTILES * 16;
    const int wave = threadIdx.x >> 5;
    const int lane = threadIdx.x & 31;
    const int tile = blockIdx.x * 4 + wave;
    if (tile >= mtiles) return;          // wave-uniform: EXEC stays all-1s

    const int row0 = tile * 16;
    const int m    = lane & 15;          // A: M index / B: N index / C: col
    const int kh   = lane >> 4;          // K-half selector (ISA A/B layout)

    const float* arow = A + (size_t)(row0 + m) * K + 2 * kh;

    v8f acc[NTILES] = {};

    for (int k = 0; k < K; k += 4) {
        float a0 = arow[k];
        float a1 = arow[k + 1];
        if (RELU) { a0 = fmaxf(a0, 0.0f); a1 = fmaxf(a1, 0.0f); }
        v2f af; af.x = a0; af.y = a1;

        const float* wk = W + (size_t)(k + 2 * kh) * N + m;
#pragma unroll
        for (int nt = 0; nt < NTILES; ++nt) {
            v2f bf; bf.x = wk[nt * 16]; bf.y = wk[N + nt * 16];
            acc[nt] = __builtin_amdgcn_wmma_f32_16x16x4_f32(
                /*neg_a=*/false, af, /*neg_b=*/false, bf,
                /*c_mod=*/(short)0, acc[nt],
                /*reuse_a=*/false, /*reuse_b=*/false);
        }
    }

#pragma unroll
    for (int nt = 0; nt < NTILES; ++nt) {
        const int col = nt * 16 + m;
        const float bv = BIAS ? bias[col] : 0.0f;
#pragma unroll
        for (int r = 0; r < 8; ++r) {
            const int row = row0 + r + 8 * kh;   // C/D VGPR layout
            C[(size_t)row * N + col] = acc[nt][r] + bv;
        }
    }
}

// ---------------------------------------------------------------------------
// GCN aggregation: out = b + self-loop + scatter-add of normalized messages
// ---------------------------------------------------------------------------
__global__ void k_agg_init(const float* __restrict__ h,
                           const float* __restrict__ dinv,
                           const float* __restrict__ b,
                           float* __restrict__ out)
{
    const int node = blockIdx.x;
    const int f    = threadIdx.x;
    const float di = dinv[node];
    out[(size_t)node * HD + f] = b[f] + h[(size_t)node * HD + f] * di * di;
}

__global__ void k_agg_edges(const long long* __restrict__ src,
                            const long long* __restrict__ dst,
                            const float* __restrict__ h,
                            const float* __restrict__ dinv,
                            float* __restrict__ out)
{
    const int e = blockIdx.x;
    const int f = threadIdx.x;
    const int s = (int)src[e];
    const int d = (int)dst[e];
    const float norm = dinv[s] * dinv[d];
    unsafeAtomicAdd(&out[(size_t)d * HD + f], h[(size_t)s * HD + f] * norm);
}

// ---------------------------------------------------------------------------
// Launch
// ---------------------------------------------------------------------------
extern "C" void kernel_launch(void* const* d_in, const int* in_sizes, int n_in,
                              void* d_out, int out_size, void* d_ws, size_t ws_size,
                              hipStream_t stream)
{
    const float*     user = (const float*)d_in[0];
    const float*     prod = (const float*)d_in[1];
    const long long* eidx = (const long long*)d_in[2];   // int64, shape [2, E]
    const float* Wu = (const float*)d_in[3];
    const float* bu = (const float*)d_in[4];
    const float* Wp = (const float*)d_in[5];
    const float* bp = (const float*)d_in[6];
    const float* W1 = (const float*)d_in[7];
    const float* b1 = (const float*)d_in[8];
    const float* W2 = (const float*)d_in[9];
    const float* b2 = (const float*)d_in[10];
    const float* W3 = (const float*)d_in[11];
    const float* b3 = (const float*)d_in[12];
    const float* Wo = (const float*)d_in[13];
    const float* bo = (const float*)d_in[14];
    float* out = (float*)d_out;

    const long long* src = eidx;
    const long long* dst = eidx + N_EDGES;

    float* ws   = (float*)d_ws;
    float* xA   = ws;                             // [N_NODES, HD]
    float* hB   = xA + (size_t)N_NODES * HD;      // [N_NODES, HD]
    float* deg  = hB + (size_t)N_NODES * HD;      // [N_NODES]
    float* dinv = deg + N_NODES;                  // [N_NODES]

    const int MT_HALF = N_USERS / 16;   // 3125
    const int MT_ALL  = N_NODES / 16;   // 6250
    dim3 blk(128);

    // --- symmetric normalization factors ---
    k_init_deg <<<(N_NODES + 255) / 256, 256, 0, stream>>>(deg);
    k_deg_edges<<<(N_EDGES + 255) / 256, 256, 0, stream>>>(dst, deg);
    k_dinv     <<<(N_NODES + 255) / 256, 256, 0, stream>>>(dinv, deg);

    // --- input projections into concatenated x ---
    gemm_wmma<8, false, true><<<(MT_HALF + 3) / 4, blk, 0, stream>>>(
        user, Wu, bu, xA, UD, MT_HALF);
    gemm_wmma<8, false, true><<<(MT_HALF + 3) / 4, blk, 0, stream>>>(
        prod, Wp, bp, xA + (size_t)N_USERS * HD, PD, MT_HALF);

    // --- GCN layer 1 ---
    gemm_wmma<8, false, false><<<(MT_ALL + 3) / 4, blk, 0, stream>>>(
        xA, W1, nullptr, hB, HD, MT_ALL);
    k_agg_init <<<N_NODES, HD, 0, stream>>>(hB, dinv, b1, xA);
    k_agg_edges<<<N_EDGES, HD, 0, stream>>>(src, dst, hB, dinv, xA);

    // --- GCN layer 2 (relu fused into A loads) ---
    gemm_wmma<8, true, false><<<(MT_ALL + 3) / 4, blk, 0, stream>>>(
        xA, W2, nullptr, hB, HD, MT_ALL);
    k_agg_init <<<N_NODES, HD, 0, stream>>>(hB, dinv, b2, xA);
    k_agg_edges<<<N_EDGES, HD, 0, stream>>>(src, dst, hB, dinv, xA);

    // --- GCN layer 3 ---
    gemm_wmma<8, true, false><<<(MT_ALL + 3) / 4, blk, 0, stream>>>(
        xA, W3, nullptr, hB, HD, MT_ALL);
    k_agg_init <<<N_NODES, HD, 0, stream>>>(hB, dinv, b3, xA);
    k_agg_edges<<<N_EDGES, HD, 0, stream>>>(src, dst, hB, dinv, xA);

    // --- output head: relu(x) @ Wo + bo -> [N_NODES, 16] ---
    gemm_wmma<1, true, true><<<(MT_ALL + 3) / 4, blk, 0, stream>>>(
        xA, Wo, bo, out, HD, MT_ALL);
}